// SepModel_80831284510852
// MI455X (gfx1250) — compile-verified
//
#include <hip/hip_runtime.h>
#include <cfloat>

typedef __attribute__((ext_vector_type(2))) float v2f;
typedef __attribute__((ext_vector_type(8))) float v8f;

#define T_DIM 512
#define B_DIM 16
#define D_DIM 1024
#define WINR  16
#define BAND  48      // 3 * 16: k-band covered by a 16-row t-tile
#define PSTR  49      // LDS row stride (odd -> conflict-free column access)

__device__ __forceinline__ int clampi(int v, int lo, int hi) {
  return v < lo ? lo : (v > hi ? hi : v);
}

// Phase 1: c[t,b] = dot(x[t,b,:], w[D:2D]).  One wave per (t,b); 8 waves/block.
__global__ __launch_bounds__(256) void seq_proj_kernel(
    const float* __restrict__ x, const float* __restrict__ w,
    float* __restrict__ c) {
  const int wv   = blockIdx.x * 8 + (threadIdx.x >> 5);  // flat t*B + b
  const int lane = threadIdx.x & 31;
  const float4* xr = reinterpret_cast<const float4*>(x) + (size_t)wv * (D_DIM / 4);
  const float4* ws = reinterpret_cast<const float4*>(w + D_DIM);
  float acc = 0.f;
#pragma unroll
  for (int q = lane; q < D_DIM / 4; q += 32) {
    float4 xv = xr[q];
    float4 wq = ws[q];
    acc = fmaf(xv.x, wq.x, acc);
    acc = fmaf(xv.y, wq.y, acc);
    acc = fmaf(xv.z, wq.z, acc);
    acc = fmaf(xv.w, wq.w, acc);
  }
#pragma unroll
  for (int off = 16; off > 0; off >>= 1) acc += __shfl_xor(acc, off, 32);
  if (lane == 0) c[wv] = acc;
}

// Phase 2: per (t-tile, b) block: softmax band P (16x48) in LDS, copy x -> out[:, :D],
// then attn = P x X via v_wmma_f32_16x16x4_f32, 12 K-steps per 16x16 output tile.
__global__ __launch_bounds__(256) void attn_wmma_kernel(
    const float* __restrict__ x, const float* __restrict__ c,
    float* __restrict__ out) {
  __shared__ float P[16 * PSTR];
  const int b    = blockIdx.x & (B_DIM - 1);
  const int t0   = (blockIdx.x >> 4) * 16;
  const int tid  = threadIdx.x;
  const int lane = tid & 31;
  const int wave = tid >> 5;

  // --- softmax rows: thread i handles row t0+i (a[t,b] and b0 cancel along w) ---
  if (tid < 16) {
    const int i = tid, t = t0 + i;
    float vals[33];
    float m = -FLT_MAX;
#pragma unroll
    for (int o = 0; o < 33; ++o) {
      const int  k  = t - WINR + o;
      const bool ok = (k >= 0) && (k < T_DIM);
      const float cv = ok ? c[k * B_DIM + b] : -FLT_MAX;
      vals[o] = cv;
      m = fmaxf(m, cv);
    }
    float s = 0.f;
#pragma unroll
    for (int o = 0; o < 33; ++o) {
      const int  k  = t - WINR + o;
      const bool ok = (k >= 0) && (k < T_DIM);
      const float e = ok ? __expf(vals[o] - m) : 0.f;
      vals[o] = e;
      s += e;
    }
    const float inv = 1.f / s;
    for (int j = 0; j < BAND; ++j) P[i * PSTR + j] = 0.f;
#pragma unroll
    for (int o = 0; o < 33; ++o) P[i * PSTR + i + o] = vals[o] * inv;
  }

  // --- pass-through half: out[t, b, 0:D] = x[t, b, :]  (float4, coalesced) ---
  {
    const float4* src = reinterpret_cast<const float4*>(x);
    float4*       dst = reinterpret_cast<float4*>(out);
    for (int e = tid; e < 16 * (D_DIM / 4); e += 256) {
      const int    i    = e >> 8;          // D_DIM/4 == 256 float4 per row
      const int    d4   = e & 255;
      const size_t srow = ((size_t)(t0 + i) * B_DIM + b) * (D_DIM / 4);
      const size_t drow = ((size_t)(t0 + i) * B_DIM + b) * (2 * D_DIM / 4);
      dst[drow + d4] = src[srow + d4];
    }
  }
  __syncthreads();

  // --- WMMA: A/B element K = j + 2*(lane>=16); C/D row M = v + 8*(lane>=16) ---
  const int row = lane & 15;
  const int hi2 = (lane >> 4) << 1;
  const int n   = lane & 15;

  v2f afr[12];
#pragma unroll
  for (int kk = 0; kk < 12; ++kk) {
    afr[kk].x = P[row * PSTR + kk * 4 + 0 + hi2];
    afr[kk].y = P[row * PSTR + kk * 4 + 1 + hi2];
  }

  for (int dt = wave; dt < D_DIM / 16; dt += 8) {
    const int d0 = dt * 16;
    v8f acc = {};
#pragma unroll
    for (int kk = 0; kk < 12; ++kk) {
      const int kb = t0 - WINR + kk * 4 + hi2;
      const int k0 = clampi(kb, 0, T_DIM - 1);      // clamp: P=0 kills invalid cols
      const int k1 = clampi(kb + 1, 0, T_DIM - 1);
      v2f bfr;
      bfr.x = x[((size_t)k0 * B_DIM + b) * D_DIM + d0 + n];
      bfr.y = x[((size_t)k1 * B_DIM + b) * D_DIM + d0 + n];
      acc = __builtin_amdgcn_wmma_f32_16x16x4_f32(
          false, afr[kk], false, bfr, (short)0, acc, false, false);
    }
    const size_t obase = (size_t)b * (2 * D_DIM) + D_DIM + d0 + n;
#pragma unroll
    for (int v = 0; v < 8; ++v) {
      const int mrow = v + ((lane >> 4) << 3);
      out[(size_t)(t0 + mrow) * B_DIM * (2 * D_DIM) + obase] = acc[v];
    }
  }
}

extern "C" void kernel_launch(void* const* d_in, const int* in_sizes, int n_in,
                              void* d_out, int out_size, void* d_ws, size_t ws_size,
                              hipStream_t stream) {
  const float* x = (const float*)d_in[0];
  const float* w = (const float*)d_in[1];  // [0:D] = w_cur (cancels), [D:2D] = w_seq
  float* c   = (float*)d_ws;               // T*B floats = 32 KB scratch
  float* out = (float*)d_out;

  seq_proj_kernel<<<(T_DIM * B_DIM) / 8, 256, 0, stream>>>(x, w, c);
  attn_wmma_kernel<<<(T_DIM / 16) * B_DIM, 256, 0, stream>>>(x, c, out);
}